// WaveletBranch_65077344469138
// MI455X (gfx1250) — compile-verified
//
#include <hip/hip_runtime.h>

typedef float v2f __attribute__((ext_vector_type(2)));
typedef float v8f __attribute__((ext_vector_type(8)));

constexpr int Bb = 4, Hh = 128, Ww = 128, Cc = 128;
constexpr int Ll = Hh * Ww;        // 16384
constexpr int ROWS = Bb * Ll;      // 65536
constexpr int XDP = 32;            // padded x_dbl row pitch (24 used)

// ---------------------------------------------------------------------------
// CDNA5 async global->LDS copy (ASYNCcnt path) + wait
// ---------------------------------------------------------------------------
__device__ __forceinline__ void async_b128_to_lds(const float* gaddr, const float* lds_dst) {
  // low 32 bits of a generic pointer to __shared__ == wave-relative LDS byte address
  unsigned lds_off = (unsigned)(size_t)lds_dst;
  asm volatile("global_load_async_to_lds_b128 %0, %1, off"
               :: "v"(lds_off), "v"(gaddr) : "memory");
}
__device__ __forceinline__ void wait_asynccnt0() {
  asm volatile("s_wait_asynccnt 0x0" ::: "memory");
}

// ---------------------------------------------------------------------------
// Generic fp32 WMMA GEMM:  out = A(MxK) * W(NxK)^T  (+ epilogues)
// Each wave computes a 16 x (NTW*16) strip: one A-frag feeds NTW WMMAs per k-step.
// MASKB=false: all W rows valid -> raw loads, zero masking VALU in the hot loop.
// MODE 0: split store: cols [0,nsplit) -> out0, cols [nsplit,N) -> out1
// MODE 1: store cols < nreal to out0 (pitch ldc)
// MODE 2: out0 = val + bias[n] + resid[m*ldc+n]
// ---------------------------------------------------------------------------
template <int NTW, int MODE, bool MASKB>
__global__ __launch_bounds__(128)
void wmma_gemm_kernel(const float* __restrict__ A, const float* __restrict__ Wt,
                      int K, int lda, int ldw, int wrows, int ntiles,
                      float* __restrict__ out0, int ldc,
                      const float* __restrict__ bias, const float* __restrict__ resid,
                      float* __restrict__ out1, int nsplit, int nreal) {
  const int wave = threadIdx.x >> 5;
  const int lane = threadIdx.x & 31;
  const int groups = ntiles / NTW;
  const int tile = blockIdx.x * 4 + wave;
  const int mt = tile / groups, ng = tile % groups;
  const int m0 = mt * 16, n0 = ng * (NTW * 16);
  const int lr = lane & 15, half = lane >> 4;

  v8f acc[NTW];
  for (int s = 0; s < NTW; ++s) acc[s] = v8f{0.f, 0.f, 0.f, 0.f, 0.f, 0.f, 0.f, 0.f};

  // A frag: lanes 0-15 rows M=lr, K={k0,k0+1}; lanes 16-31 rows M=lr, K={k0+2,k0+3}
  const float* arow = A + (size_t)(m0 + lr) * lda + 2 * half;
  const float* wrow[NTW];
  bool wok[NTW];
  for (int s = 0; s < NTW; ++s) {
    const int nW = n0 + s * 16 + lr;
    wok[s] = (nW < wrows);
    const int nc = (MASKB && !wok[s]) ? (wrows - 1) : nW;  // clamp only when masking
    wrow[s] = Wt + (size_t)nc * ldw + 2 * half;
  }
  for (int k0 = 0; k0 < K; k0 += 4) {
    v2f a;
    a.x = arow[k0];
    a.y = arow[k0 + 1];
    for (int s = 0; s < NTW; ++s) {
      v2f b;
      const float bx = wrow[s][k0];
      const float by = wrow[s][k0 + 1];
      if (MASKB) {
        b.x = wok[s] ? bx : 0.f;                // v_cndmask, no exec divergence
        b.y = wok[s] ? by : 0.f;
      } else {
        b.x = bx;
        b.y = by;
      }
      acc[s] = __builtin_amdgcn_wmma_f32_16x16x4_f32(false, a, false, b, (short)0,
                                                     acc[s], false, false);
    }
  }
  for (int s = 0; s < NTW; ++s) {
    const int n = n0 + s * 16 + lr;
    for (int v = 0; v < 8; ++v) {
      const int m = m0 + v + 8 * half;
      const float val = acc[s][v];
      if (MODE == 0) {
        if (n < nsplit) out0[(size_t)m * nsplit + n] = val;
        else            out1[(size_t)m * nsplit + (n - nsplit)] = val;
      } else if (MODE == 1) {
        if (n < nreal) out0[(size_t)m * ldc + n] = val;
      } else {
        out0[(size_t)m * ldc + n] = val + bias[n] + resid[(size_t)m * ldc + n];
      }
    }
  }
}

// ---------------------------------------------------------------------------
// Haar DWT level 1 (channel-last): x_proc(B,L,C) -> Yl1(B,64,64,C), bands1(B,64,64,3,C)
// ---------------------------------------------------------------------------
__global__ void dwt1_kernel(const float* __restrict__ xp, float* __restrict__ yl1,
                            float* __restrict__ bands1) {
  const int b = blockIdx.y;
  const int i1 = blockIdx.x >> 6, j1 = blockIdx.x & 63;
  const int c = threadIdx.x;
  const size_t base = ((size_t)b * Ll + (size_t)(2 * i1) * Ww + 2 * j1) * Cc + c;
  const float av = xp[base];
  const float bv = xp[base + Cc];
  const float cv = xp[base + (size_t)Ww * Cc];
  const float dv = xp[base + (size_t)(Ww + 1) * Cc];
  const float ll = (av + bv + cv + dv) * 0.5f;
  const float lh = (av + bv - cv - dv) * 0.5f;
  const float hl = (av - bv + cv - dv) * 0.5f;
  const float hh = (av - bv - cv + dv) * 0.5f;
  const size_t p = (size_t)b * 4096 + i1 * 64 + j1;
  yl1[p * Cc + c] = ll;
  bands1[(p * 3 + 0) * Cc + c] = lh;
  bands1[(p * 3 + 1) * Cc + c] = hl;
  bands1[(p * 3 + 2) * Cc + c] = hh;
}

// ---------------------------------------------------------------------------
// Haar DWT level 2 + scatter into h00_seq (B,L,C). h00 must be pre-zeroed.
// ---------------------------------------------------------------------------
__global__ void dwt2_scatter_kernel(const float* __restrict__ yl1, float* __restrict__ h00) {
  const int b = blockIdx.y;
  const int i2 = blockIdx.x >> 5, j2 = blockIdx.x & 31;
  const int c = threadIdx.x;
  const size_t r0 = ((size_t)b * 4096 + (size_t)(2 * i2) * 64 + 2 * j2) * Cc + c;
  const float av = yl1[r0];
  const float bv = yl1[r0 + Cc];
  const float cv = yl1[r0 + (size_t)64 * Cc];
  const float dv = yl1[r0 + (size_t)65 * Cc];
  const float ll = (av + bv + cv + dv) * 0.5f;
  const float lh = (av + bv - cv - dv) * 0.5f;
  const float hl = (av - bv + cv - dv) * 0.5f;
  const float hh = (av - bv - cv + dv) * 0.5f;
  const size_t ib = (size_t)b * Ll;
  h00[(ib + (size_t)i2 * Ww + j2) * Cc + c] = ll;
  h00[(ib + (size_t)i2 * Ww + j2 + 32) * Cc + c] = lh;
  h00[(ib + (size_t)(i2 + 32) * Ww + j2) * Cc + c] = hl;
  h00[(ib + (size_t)(i2 + 32) * Ww + j2 + 32) * Cc + c] = hh;
}

// ---------------------------------------------------------------------------
// Tiled transpose per batch: in(B, R, Ccols) -> out(B, Ccols, R)
// ---------------------------------------------------------------------------
__global__ void transpose_kernel(const float* __restrict__ in, float* __restrict__ out,
                                 int R, int Ccols) {
  __shared__ float t[32][33];
  const int b = blockIdx.z;
  const int r0 = blockIdx.x * 32, c0 = blockIdx.y * 32;
  const float* ib = in + (size_t)b * R * Ccols;
  float* ob = out + (size_t)b * R * Ccols;
  for (int i = 0; i < 4; ++i) {
    const int r = r0 + threadIdx.y + i * 8;
    t[threadIdx.y + i * 8][threadIdx.x] = ib[(size_t)r * Ccols + c0 + threadIdx.x];
  }
  __syncthreads();
  for (int i = 0; i < 4; ++i) {
    const int cidx = c0 + threadIdx.y + i * 8;
    ob[(size_t)cidx * R + r0 + threadIdx.x] = t[threadIdx.x][threadIdx.y + i * 8];
  }
}

// ---------------------------------------------------------------------------
// Selective scan. One wave32 per (b,c) row; 4 waves/block share one b.
// Chunked associative scan: per chunk of CH, each lane folds a SEG=16 segment
// into (P,S); 5-step shfl_up scan composes lane carries; pass B emits y.
// delta = softplus(dts . dt_proj_w[c,:] + dt_proj_b[c]) fused (no delta buffer).
// Staging via GLOBAL_LOAD_ASYNC_TO_LDS_B128 (ASYNCcnt); next chunk prefetched.
// LDS strides: x_dbl rows 36 floats (144B: 16B-aligned, 2-way-conflict reads),
// u/y rows 20 floats (80B).
// ---------------------------------------------------------------------------
constexpr int CH = 512;
constexpr int SEG = CH / 32;   // 16
constexpr int NCHUNK = Ll / CH;

__device__ __forceinline__ float softplus_f(float x) {
  return (x > 20.f) ? x : log1pf(__expf(x));
}

__global__ __launch_bounds__(128)
void scan_kernel(const float* __restrict__ ut, const float* __restrict__ xdbl,
                 const float* __restrict__ prompt, const float* __restrict__ dtw,
                 const float* __restrict__ dtb, const float* __restrict__ Alogs,
                 const float* __restrict__ Dsv, float* __restrict__ yt) {
  __shared__ __align__(16) float s_xd[SEG * 32 * 36];  // [t][lane][36], k=0..31 of x_dbl rows
  __shared__ __align__(16) float s_u[4 * 32 * 20];     // [wave][lane][20]
  __shared__ __align__(16) float s_y[4 * 32 * 20];
  const int w = threadIdx.x >> 5, lane = threadIdx.x & 31, tid = threadIdx.x;
  const int bb = blockIdx.x >> 5;
  const int cg = blockIdx.x & 31;
  const int c = cg * 4 + w;

  float dw[8], An[8], hc[8];
  for (int r = 0; r < 8; ++r) dw[r] = dtw[c * 8 + r];
  const float bias = dtb[c];
  for (int n = 0; n < 8; ++n) An[n] = -__expf(Alogs[c * 8 + n]);
  const float Dc = Dsv[c];
  for (int n = 0; n < 8; ++n) hc[n] = 0.f;

  const float* urow = ut + (size_t)(bb * Cc + c) * Ll;
  float* yrow = yt + (size_t)(bb * Cc + c) * Ll;
  const float* xrow = xdbl + (size_t)bb * Ll * XDP;
  const float* prow = prompt + (size_t)bb * Ll * 8;

  for (int ch = 0; ch < NCHUNK; ++ch) {
    const int l0 = ch * CH;
    // ---- async stage x_dbl rows (all 32 dwords/row; shared by the 4 waves, same b)
    for (int i = tid; i < CH * 8; i += 128) {
      const int li = i >> 3, q = i & 7;
      async_b128_to_lds(xrow + (size_t)(l0 + li) * XDP + q * 4,
                        &s_xd[((li & 15) * 32 + (li >> 4)) * 36 + q * 4]);
    }
    // ---- async stage this wave's u segment
    for (int j = lane; j < CH / 4; j += 32) {
      const int li = j * 4;
      async_b128_to_lds(urow + l0 + li,
                        &s_u[(w * 32 + (li >> 4)) * 20 + (li & 15)]);
    }
    wait_asynccnt0();
    __syncthreads();

    // ---- prefetch next chunk while computing this one
    if (ch + 1 < NCHUNK) {
      const float* nx = xrow + (size_t)(l0 + CH) * XDP;
      for (int i = 0; i < 4; ++i)
        __builtin_prefetch(nx + (size_t)(tid + i * 128) * 32, 0, 0);
      if (lane < 16) __builtin_prefetch(urow + l0 + CH + lane * 32, 0, 0);
    }

    // ---- pass A: per-lane segment fold into (P, S)
    float P[8], S[8];
    for (int n = 0; n < 8; ++n) { P[n] = 1.f; S[n] = 0.f; }
    for (int t = 0; t < SEG; ++t) {
      const float* xd = &s_xd[(t * 32 + lane) * 36];
      float dval = bias;
      for (int r = 0; r < 8; ++r) dval += xd[r] * dw[r];
      const float delta = softplus_f(dval);
      const float uv = s_u[(w * 32 + lane) * 20 + t];
      const float du = delta * uv;
      for (int n = 0; n < 8; ++n) {
        const float a = __expf(delta * An[n]);
        S[n] = a * S[n] + du * xd[8 + n];
        P[n] *= a;
      }
    }
    // ---- wave inclusive scan over lane aggregates
    for (int d = 1; d < 32; d <<= 1) {
      for (int n = 0; n < 8; ++n) {
        const float Pp = __shfl_up(P[n], d, 32);
        const float Sp = __shfl_up(S[n], d, 32);
        if (lane >= d) { S[n] = P[n] * Sp + S[n]; P[n] = P[n] * Pp; }
      }
    }
    float h[8];
    for (int n = 0; n < 8; ++n) {
      float Pe = __shfl_up(P[n], 1, 32);
      float Se = __shfl_up(S[n], 1, 32);
      if (lane == 0) { Pe = 1.f; Se = 0.f; }
      h[n] = Pe * hc[n] + Se;                    // carry-in for this lane's segment
      const float P31 = __shfl(P[n], 31, 32);
      const float S31 = __shfl(S[n], 31, 32);
      hc[n] = P31 * hc[n] + S31;                 // chunk carry for next chunk
    }
    // ---- pass B: replay segment with correct carry, emit y
    for (int t = 0; t < SEG; ++t) {
      const int li = lane * SEG + t;
      const float* xd = &s_xd[(t * 32 + lane) * 36];
      float dval = bias;
      for (int r = 0; r < 8; ++r) dval += xd[r] * dw[r];
      const float delta = softplus_f(dval);
      const float uv = s_u[(w * 32 + lane) * 20 + t];
      const float du = delta * uv;
      const float* pp = prow + (size_t)(l0 + li) * 8;
      float y = 0.f;
      for (int n = 0; n < 8; ++n) {
        const float a = __expf(delta * An[n]);
        h[n] = a * h[n] + du * xd[8 + n];
        y += h[n] * (xd[16 + n] + pp[n]);
      }
      y += uv * Dc;
      s_y[(w * 32 + lane) * 20 + t] = y;
    }
    // coalesced y writeback (LDS ops are in-order within a wave)
    for (int j = lane; j < CH / 4; j += 32) {
      const int li = j * 4;
      const int base = (w * 32 + (li >> 4)) * 20 + (li & 15);
      float4 v;
      v.x = s_y[base + 0]; v.y = s_y[base + 1]; v.z = s_y[base + 2]; v.w = s_y[base + 3];
      *reinterpret_cast<float4*>(yrow + l0 + li) = v;
    }
    __syncthreads();
  }
}

// ---------------------------------------------------------------------------
// Inverse DWT level 2 (channel-last): h11_seq(B,L,C) quadrants -> Yl1n(B,64,64,C)
// ---------------------------------------------------------------------------
__global__ void idwt2_kernel(const float* __restrict__ h11, float* __restrict__ yl1n) {
  const int b = blockIdx.y;
  const int i2 = blockIdx.x >> 5, j2 = blockIdx.x & 31;
  const int c = threadIdx.x;
  const size_t ib = (size_t)b * Ll;
  const float ll = h11[(ib + (size_t)i2 * Ww + j2) * Cc + c];
  const float lh = h11[(ib + (size_t)i2 * Ww + j2 + 32) * Cc + c];
  const float hl = h11[(ib + (size_t)(i2 + 32) * Ww + j2) * Cc + c];
  const float hh = h11[(ib + (size_t)(i2 + 32) * Ww + j2 + 32) * Cc + c];
  const float av = (ll + lh + hl + hh) * 0.5f;
  const float bv = (ll + lh - hl - hh) * 0.5f;
  const float cv = (ll - lh + hl - hh) * 0.5f;
  const float dv = (ll - lh - hl + hh) * 0.5f;
  const size_t pb = (size_t)b * 4096;
  yl1n[(pb + (size_t)(2 * i2) * 64 + 2 * j2) * Cc + c] = av;
  yl1n[(pb + (size_t)(2 * i2) * 64 + 2 * j2 + 1) * Cc + c] = bv;
  yl1n[(pb + (size_t)(2 * i2 + 1) * 64 + 2 * j2) * Cc + c] = cv;
  yl1n[(pb + (size_t)(2 * i2 + 1) * 64 + 2 * j2 + 1) * Cc + c] = dv;
}

// ---------------------------------------------------------------------------
// Inverse DWT level 1 (channel-last): Yl1n + bands1 -> y_seq(B,L,C)
// ---------------------------------------------------------------------------
__global__ void idwt1_kernel(const float* __restrict__ yl1n, const float* __restrict__ bands1,
                             float* __restrict__ yseq) {
  const int b = blockIdx.y;
  const int i1 = blockIdx.x >> 6, j1 = blockIdx.x & 63;
  const int c = threadIdx.x;
  const size_t p = (size_t)b * 4096 + i1 * 64 + j1;
  const float ll = yl1n[p * Cc + c];
  const float lh = bands1[(p * 3 + 0) * Cc + c];
  const float hl = bands1[(p * 3 + 1) * Cc + c];
  const float hh = bands1[(p * 3 + 2) * Cc + c];
  const float av = (ll + lh + hl + hh) * 0.5f;
  const float bv = (ll + lh - hl - hh) * 0.5f;
  const float cv = (ll - lh + hl - hh) * 0.5f;
  const float dv = (ll - lh - hl + hh) * 0.5f;
  const size_t ib = (size_t)b * Ll;
  yseq[(ib + (size_t)(2 * i1) * Ww + 2 * j1) * Cc + c] = av;
  yseq[(ib + (size_t)(2 * i1) * Ww + 2 * j1 + 1) * Cc + c] = bv;
  yseq[(ib + (size_t)(2 * i1 + 1) * Ww + 2 * j1) * Cc + c] = cv;
  yseq[(ib + (size_t)(2 * i1 + 1) * Ww + 2 * j1 + 1) * Cc + c] = dv;
}

// ---------------------------------------------------------------------------
// LayerNorm over C + SiLU gate, in place on y. One wave32 per row (4 floats/lane).
// ---------------------------------------------------------------------------
__global__ __launch_bounds__(256)
void ln_gate_kernel(float* __restrict__ y, const float* __restrict__ z,
                    const float* __restrict__ gamma, const float* __restrict__ beta) {
  const int row = blockIdx.x * 8 + (threadIdx.x >> 5);
  const int lane = threadIdx.x & 31;
  float* yr = y + (size_t)row * Cc;
  const float* zr = z + (size_t)row * Cc;
  const float4 v = *reinterpret_cast<const float4*>(yr + lane * 4);
  float s = v.x + v.y + v.z + v.w;
  float sq = v.x * v.x + v.y * v.y + v.z * v.z + v.w * v.w;
  for (int d = 16; d > 0; d >>= 1) {
    s += __shfl_xor(s, d, 32);
    sq += __shfl_xor(sq, d, 32);
  }
  const float mean = s * (1.f / 128.f);
  const float var = sq * (1.f / 128.f) - mean * mean;
  const float rs = rsqrtf(var + 1e-5f);
  const float4 g = *reinterpret_cast<const float4*>(gamma + lane * 4);
  const float4 be = *reinterpret_cast<const float4*>(beta + lane * 4);
  const float4 zv = *reinterpret_cast<const float4*>(zr + lane * 4);
  float4 o;
  o.x = ((v.x - mean) * rs * g.x + be.x) * (zv.x / (1.f + __expf(-zv.x)));
  o.y = ((v.y - mean) * rs * g.y + be.y) * (zv.y / (1.f + __expf(-zv.y)));
  o.z = ((v.z - mean) * rs * g.z + be.z) * (zv.z / (1.f + __expf(-zv.z)));
  o.w = ((v.w - mean) * rs * g.w + be.w) * (zv.w / (1.f + __expf(-zv.w)));
  *reinterpret_cast<float4*>(yr + lane * 4) = o;
}

// ---------------------------------------------------------------------------
extern "C" void kernel_launch(void* const* d_in, const int* in_sizes, int n_in,
                              void* d_out, int out_size, void* d_ws, size_t ws_size,
                              hipStream_t stream) {
  const float* x         = (const float*)d_in[0];
  const float* prompt    = (const float*)d_in[1];
  const float* in_proj_w = (const float*)d_in[2];
  const float* x_proj_w  = (const float*)d_in[3];
  const float* dt_proj_w = (const float*)d_in[4];
  const float* dt_proj_b = (const float*)d_in[5];
  const float* A_logs    = (const float*)d_in[6];
  const float* Ds        = (const float*)d_in[7];
  const float* ln_gamma  = (const float*)d_in[8];
  const float* ln_beta   = (const float*)d_in[9];
  const float* out_proj_w = (const float*)d_in[10];
  const float* out_proj_b = (const float*)d_in[11];
  float* out = (float*)d_out;

  // workspace layout (floats); buffers reused across pipeline phases
  float* ws = (float*)d_ws;
  const size_t S = (size_t)ROWS * Cc;                 // 8,388,608
  float* x_proc = ws;                                 // later: h11_seq
  float* z      = ws + S;
  float* h00    = ws + 2 * S;                         // later: y_t (B,C,L)
  float* u_t    = ws + 3 * S;                         // later: y_seq / y_gated
  float* x_dbl  = ws + 4 * S;                         // ROWS*32
  float* yl1    = x_dbl + (size_t)ROWS * XDP;         // B*4096*C; later: Yl1n
  float* bands1 = yl1 + (size_t)Bb * 4096 * Cc;       // B*4096*3*C
  // total: 4*S + 2*2.1M + 6.3M floats ~= 176 MB

  // 1) in_proj: xz = x @ in_proj_w^T, split into x_proc / z (no B-masking needed)
  wmma_gemm_kernel<4, 0, false><<<4096, 128, 0, stream>>>(
      x, in_proj_w, 128, 128, 128, 256, 16, x_proc, 128, nullptr, nullptr, z, 128, 0);
  // 2) DWT level 1
  dwt1_kernel<<<dim3(4096, Bb), 128, 0, stream>>>(x_proc, yl1, bands1);
  // 3) zero h00_seq, then DWT level 2 + quadrant scatter
  hipMemsetAsync(h00, 0, S * sizeof(float), stream);
  dwt2_scatter_kernel<<<dim3(1024, Bb), 128, 0, stream>>>(yl1, h00);
  // 4) x_dbl = h00_seq @ x_proj_w^T (24 real cols, padded pitch 32); masked B rows
  wmma_gemm_kernel<2, 1, true><<<1024, 128, 0, stream>>>(
      h00, x_proj_w, 128, 128, 128, 24, 2, x_dbl, 32, nullptr, nullptr, nullptr, 0, 24);
  // 5) transpose u: h00_seq (B,L,C) -> u_t (B,C,L)
  transpose_kernel<<<dim3(Ll / 32, Cc / 32, Bb), dim3(32, 8), 0, stream>>>(h00, u_t, Ll, Cc);
  // 6) selective scan (delta fused); writes y_t into h00's region (B,C,L)
  scan_kernel<<<128, 128, 0, stream>>>(u_t, x_dbl, prompt, dt_proj_w, dt_proj_b,
                                       A_logs, Ds, h00);
  // 7) transpose back: y_t (B,C,L) -> h11_seq (B,L,C) into x_proc region
  transpose_kernel<<<dim3(Cc / 32, Ll / 32, Bb), dim3(32, 8), 0, stream>>>(h00, x_proc, Cc, Ll);
  // 8) inverse DWT level 2 -> Yl1n (reuse yl1 region)
  idwt2_kernel<<<dim3(1024, Bb), 128, 0, stream>>>(x_proc, yl1);
  // 9) inverse DWT level 1 -> y_seq (reuse u_t region)
  idwt1_kernel<<<dim3(4096, Bb), 128, 0, stream>>>(yl1, bands1, u_t);
  // 10) layernorm + SiLU(z) gate, in place
  ln_gate_kernel<<<ROWS / 8, 256, 0, stream>>>(u_t, z, ln_gamma, ln_beta);
  // 11) out_proj + bias + residual x -> d_out (no B-masking needed)
  wmma_gemm_kernel<4, 2, false><<<2048, 128, 0, stream>>>(
      u_t, out_proj_w, 128, 128, 128, 128, 8, out, 128, out_proj_b, x, nullptr, 0, 128);
}